// OutputLayer_18786186953532
// MI455X (gfx1250) — compile-verified
//
#include <hip/hip_runtime.h>

// F is fixed at 128 by the reference problem.
#define FDIM 128

typedef __attribute__((ext_vector_type(2))) float v2f;
typedef __attribute__((ext_vector_type(8))) float v8f;

__global__ void zero_out_kernel(float* __restrict__ out, int M) {
    int i = blockIdx.x * blockDim.x + threadIdx.x;
    if (i < M) out[i] = 0.0f;
}

// Ws = W + W^T  (128x128)
__global__ void symmetrize_kernel(const float* __restrict__ W, float* __restrict__ Ws) {
    int i = blockIdx.x * blockDim.x + threadIdx.x;   // 0..16383
    int r = i >> 7, c = i & 127;
    Ws[i] = W[r * FDIM + c] + W[c * FDIM + r];
}

// Y[N,128] = X[N,128] @ Ws[128,128] using V_WMMA_F32_16X16X4_F32 (wave32).
// Block = 256 threads = 8 waves. Block b owns rows [16b,16b+16); wave w owns cols [16w,16w+16).
__global__ void __launch_bounds__(256)
gemm_xw_kernel(const float* __restrict__ X, const float* __restrict__ Ws,
               float* __restrict__ Y, int N) {
    __shared__ float sW[FDIM * FDIM];   // 64 KB of the 320 KB/WGP LDS
    const int tid = threadIdx.x;
    {
        const float4* src = (const float4*)Ws;
        float4* dst = (float4*)sW;
        for (int i = tid; i < FDIM * FDIM / 4; i += 256) dst[i] = src[i];
    }
    __syncthreads();

    const int wave = tid >> 5;         // column tile 0..7
    const int lane = tid & 31;
    const int half = lane >> 4;        // 0: K-pair {0,1}, 1: K-pair {2,3}
    const int mn   = lane & 15;        // M (for A) / N (for B,C,D)
    const int row0 = blockIdx.x * 16;
    const int col0 = wave * 16;

    int arow = row0 + mn;
    if (arow >= N) arow = N - 1;       // clamp; stores are guarded below
    const float* A = X + (long long)arow * FDIM;

    v8f acc = {0.f, 0.f, 0.f, 0.f, 0.f, 0.f, 0.f, 0.f};
    for (int kb = 0; kb < FDIM; kb += 4) {
        v2f a, b;
        // A 16x4 layout: VGPR0/1 hold K = 2*half, 2*half+1 for row mn (contiguous pair -> b64 load)
        a.x = A[kb + 2 * half];
        a.y = A[kb + 2 * half + 1];
        // B 4x16 layout: row-striped across lanes; VGPR0/1 hold K = 2*half, 2*half+1 at col mn
        b.x = sW[(kb + 2 * half) * FDIM + col0 + mn];
        b.y = sW[(kb + 2 * half + 1) * FDIM + col0 + mn];
        acc = __builtin_amdgcn_wmma_f32_16x16x4_f32(false, a, false, b,
                                                    (short)0, acc, false, false);
    }
    // C/D 16x16 f32 layout: VGPR v -> M = v + 8*half, N = mn
#pragma unroll
    for (int v = 0; v < 8; v++) {
        int m = row0 + v + 8 * half;
        if (m < N) Y[(long long)m * FDIM + col0 + mn] = acc[v];
    }
}

// One wave (32 lanes) per edge: s = Y[i] . X[j]; LDS-aggregated segment sum.
__global__ void __launch_bounds__(256)
edge_kernel(const float* __restrict__ Y, const float* __restrict__ X,
            const long long* __restrict__ nbr, const long long* __restrict__ mid,
            float* __restrict__ out, long long E, int M) {
    __shared__ float lacc[1024];
    const bool useLds = (M <= 1024);
    const int tid = threadIdx.x;
    if (useLds) {
        for (int i = tid; i < M; i += 256) lacc[i] = 0.0f;
        __syncthreads();
    }
    const int lane = tid & 31;
    long long wave   = (long long)blockIdx.x * 8 + (tid >> 5);
    long long nwaves = (long long)gridDim.x * 8;
    for (long long e = wave; e < E; e += nwaves) {
        long long i = nbr[e];
        long long j = nbr[E + e];
        const float4* yi = (const float4*)(Y + i * FDIM);
        const float4* xj = (const float4*)(X + j * FDIM);
        float4 a = yi[lane];     // coalesced b128, L2-resident
        float4 b = xj[lane];
        float s = a.x * b.x + a.y * b.y + a.z * b.z + a.w * b.w;
#pragma unroll
        for (int o = 16; o > 0; o >>= 1) s += __shfl_xor(s, o, 32);   // wave32 reduce
        if (lane == 0) {
            int m = (int)mid[e];
            if (useLds) atomicAdd(&lacc[m], s);
            else        atomicAdd(&out[m], s);
        }
    }
    if (useLds) {
        __syncthreads();
        for (int i = tid; i < M; i += 256) {
            float v = lacc[i];
            if (v != 0.0f) atomicAdd(&out[i], v);
        }
    }
}

// Fallback if workspace cannot hold Y: one thread per edge, direct bilinear form.
__global__ void direct_edge_kernel(const float* __restrict__ Xf, const float* __restrict__ W,
                                   const long long* __restrict__ nbr,
                                   const long long* __restrict__ mid,
                                   float* __restrict__ out, long long E) {
    long long e = (long long)blockIdx.x * blockDim.x + threadIdx.x;
    if (e >= E) return;
    const float* fi = Xf + nbr[e] * FDIM;
    const float* fj = Xf + nbr[E + e] * FDIM;
    float s = 0.f;
    for (int r = 0; r < FDIM; r++) {
        float wi = 0.f, wj = 0.f;
        for (int c = 0; c < FDIM; c++) {
            wi += W[r * FDIM + c] * fj[c];   // (W  fj)[r]
            wj += W[c * FDIM + r] * fj[c];   // (W^T fj)[r]
        }
        s += fi[r] * (wi + wj);
    }
    atomicAdd(&out[(int)mid[e]], s);
}

extern "C" void kernel_launch(void* const* d_in, const int* in_sizes, int n_in,
                              void* d_out, int out_size, void* d_ws, size_t ws_size,
                              hipStream_t stream) {
    const float*     Xf  = (const float*)d_in[0];       // [N,128] f32
    const float*     W   = (const float*)d_in[1];       // [128,128] f32
    const long long* nbr = (const long long*)d_in[2];   // [2,E] int64
    const long long* mid = (const long long*)d_in[3];   // [E] int64
    float* out = (float*)d_out;

    const long long N = in_sizes[0] / FDIM;
    const long long E = in_sizes[2] / 2;
    const int       M = out_size;

    zero_out_kernel<<<(M + 255) / 256, 256, 0, stream>>>(out, M);

    const size_t need = (size_t)(FDIM * FDIM) * sizeof(float)
                      + (size_t)N * FDIM * sizeof(float);
    if (ws_size >= need) {
        float* Ws = (float*)d_ws;
        float* Y  = Ws + FDIM * FDIM;

        symmetrize_kernel<<<(FDIM * FDIM) / 256, 256, 0, stream>>>(W, Ws);

        int gblocks = (int)((N + 15) / 16);
        gemm_xw_kernel<<<gblocks, 256, 0, stream>>>(Xf, Ws, Y, (int)N);

        edge_kernel<<<1024, 256, 0, stream>>>(Y, Xf, nbr, mid, out, E, M);
    } else {
        long long blocks = (E + 255) / 256;
        direct_edge_kernel<<<(int)blocks, 256, 0, stream>>>(Xf, W, nbr, mid, out, E);
    }
}